// ConstrainedLayer_33784212750850
// MI455X (gfx1250) — compile-verified
//
#include <hip/hip_runtime.h>

// Elementwise closed-form QP solution:
//   out[i] = clip(pred[i], min(-9*y[i], 11*y[i]), max(-9*y[i], 11*y[i]))
//
// Memory-bound: 12 B/element, ~201 MB total per call -> ~8.6 us floor at
// 23.3 TB/s HBM. Strategy: float4 (b128) non-temporal global loads/stores,
// wave32-friendly 256-thread blocks, one float4 per thread (fully coalesced,
// enough waves in flight to saturate HBM: 16384 blocks x 8 waves).

typedef __attribute__((ext_vector_type(4))) float v4f;

__global__ __launch_bounds__(256) void constrained_clip_v4(
    const v4f* __restrict__ pred,
    const v4f* __restrict__ y,
    v4f* __restrict__ out,
    int n4)
{
    const int i = blockIdx.x * blockDim.x + threadIdx.x;
    if (i >= n4) return;

    // NT policy: streaming data, working set (> L2 192MB) never reused.
    const v4f p = __builtin_nontemporal_load(&pred[i]);
    const v4f t = __builtin_nontemporal_load(&y[i]);

    const v4f a = -9.0f * t;   // bound when y > 0: lower; y < 0: upper
    const v4f b = 11.0f * t;

    v4f r;
    // fminf/fmaxf -> v_min_num_f32 / v_max_num_f32 (NaN-number semantics,
    // matches jnp.minimum/maximum/clip on non-NaN inputs).
#pragma unroll
    for (int k = 0; k < 4; ++k) {
        const float lo = fminf(a[k], b[k]);
        const float hi = fmaxf(a[k], b[k]);
        r[k] = fminf(fmaxf(p[k], lo), hi);
    }

    __builtin_nontemporal_store(r, &out[i]);
}

// Defensive scalar tail (n % 4 != 0 never happens for BATCH = 2^24, but keep
// the launch correct for arbitrary n).
__global__ __launch_bounds__(64) void constrained_clip_tail(
    const float* __restrict__ pred,
    const float* __restrict__ y,
    float* __restrict__ out,
    int start, int n)
{
    const int i = start + blockIdx.x * blockDim.x + threadIdx.x;
    if (i >= n) return;
    const float p = pred[i];
    const float t = y[i];
    const float a = -9.0f * t;
    const float b = 11.0f * t;
    const float lo = fminf(a, b);
    const float hi = fmaxf(a, b);
    out[i] = fminf(fmaxf(p, lo), hi);
}

extern "C" void kernel_launch(void* const* d_in, const int* in_sizes, int n_in,
                              void* d_out, int out_size, void* d_ws, size_t ws_size,
                              hipStream_t stream)
{
    const float* pred = (const float*)d_in[0];   // predictions, fp32
    const float* y    = (const float*)d_in[1];   // y_true_batch, fp32
    float* out        = (float*)d_out;

    const int n  = in_sizes[0];      // 16,777,216
    const int n4 = n >> 2;           // 4,194,304 float4 elements
    const int rem = n - (n4 << 2);

    if (n4 > 0) {
        const int block = 256;
        const int grid  = (n4 + block - 1) / block;  // 16384 blocks
        constrained_clip_v4<<<grid, block, 0, stream>>>(
            (const v4f*)pred, (const v4f*)y, (v4f*)out, n4);
    }
    if (rem > 0) {
        constrained_clip_tail<<<1, 64, 0, stream>>>(pred, y, out, n4 << 2, n);
    }
}